// ConvCaps_4054449127616
// MI455X (gfx1250) — compile-verified
//
#include <hip/hip_runtime.h>
#include <hip/hip_bf16.h>
#include <math.h>

// ---------------------------------------------------------------------------
// EM-routing ConvCaps for MI455X (gfx1250, wave32).
//   b=8, B=C=32, K=3, stride=2, Win=13, Wout=6, n_iter=3, pose 4x4.
// Strategy: materialize votes (170MB, fits 192MB L2) with V_WMMA_F32_16X16X4_F32
// (the q-contraction is exactly K=4), then run single-pass-moment M-steps and
// shuffle-reduced E-steps streaming votes from L2.
// ---------------------------------------------------------------------------

typedef __attribute__((ext_vector_type(2))) float v2f;
typedef __attribute__((ext_vector_type(8))) float v8f;

#define NB 8
#define BCAP 32
#define CCAP 32
#define KSZ 3
#define WIN 13
#define WOUT 6
#define NPOS (WOUT*WOUT)            // 36
#define CH_IN (16*BCAP + BCAP)      // 544
#define OIJ (BCAP*KSZ*KSZ)          // 288

// votes layout: [n][o][i][j][x][y][c][d]   d = p*4 + r
#define VOTES_N  ((size_t)NB*BCAP*KSZ*KSZ*NPOS*CCAP*16)   // 42,467,328
#define RP_N     ((size_t)NB*BCAP*KSZ*KSZ*NPOS*CCAP)      //  2,654,208
#define MU_N     ((size_t)NB*NPOS*CCAP*16)                //    147,456
#define ACT_N    ((size_t)NB*NPOS*CCAP)                   //      9,216
#define D_N      ((size_t)NB*BCAP*WIN*WIN)                //     43,264

// ---------------------------------------------------------------------------
// Kernel 1: votes via WMMA f32 16x16x4.
// block = one (n,o,i,j); 8 waves; wave g covers (c,p)-rows [16g,16g+16).
// A (16x4) = W[i,j,o, c,p, q] ; B (4x16) = pose[n,q,r,o,2x+i,2y+j] with
// cols = (r, 4 spatial positions); D (16x16) = vote tile.
// ---------------------------------------------------------------------------
__global__ __launch_bounds__(256) void cc_votes_wmma(
    const float* __restrict__ xin, const float* __restrict__ Wt,
    float* __restrict__ votes)
{
  int blk = blockIdx.x;                 // 0..2303
  int j = blk % 3;
  int i = (blk / 3) % 3;
  int o = (blk / 9) % BCAP;
  int n = blk / (9 * BCAP);

  int lane  = threadIdx.x & 31;
  int g     = threadIdx.x >> 5;         // cp-group 0..7
  int khalf = lane >> 4;                // 0: q={0,1}, 1: q={2,3}

  // ---- A operand (held for all 9 spatial groups) ----
  int mRow = g * 16 + (lane & 15);      // (c,p) row
  int cA = mRow >> 2;
  int pA = mRow & 3;
  const float* wb = Wt + ((((size_t)((i*3 + j)*BCAP + o))*CCAP + cA)*4 + pA)*4;
  v2f a;
  a.x = wb[2*khalf + 0];
  a.y = wb[2*khalf + 1];

  // ---- B column decode (fixed per lane) ----
  int col = lane & 15;
  int r = col >> 2;                     // pose minor index
  int t = col & 3;                      // spatial sub-index

  for (int xyG = 0; xyG < 9; ++xyG) {
    int xy = xyG * 4 + t;
    int xx = xy / WOUT, yy = xy % WOUT;
    int u = 2*xx + i, w = 2*yy + j;
    // pose channel = q*128 + r*32 + o
    size_t pb = ((size_t)n*CH_IN + (size_t)(2*khalf)*128 + r*32 + o)*(WIN*WIN)
                + (size_t)u*WIN + w;
    v2f bm;
    bm.x = xin[pb];
    bm.y = xin[pb + (size_t)128*(WIN*WIN)];   // q+1

    v8f acc = {};
    acc = __builtin_amdgcn_wmma_f32_16x16x4_f32(
        false, a, false, bm, (short)0, acc, false, false);

    // D layout: N = lane%16 (== col), M = vgpr + 8*(lane/16)
    #pragma unroll
    for (int vv = 0; vv < 8; ++vv) {
      int M  = vv + 8*khalf;
      int mR = g*16 + M;
      int cD = mR >> 2;
      int pD = mR & 3;
      int d  = pD*4 + r;
      size_t oidx = ((((((size_t)n*BCAP + o)*3 + i)*3 + j)*WOUT + xx)*WOUT + yy)
                    * CCAP + cD;
      votes[oidx*16 + d] = acc[vv];
    }
  }
}

// ---------------------------------------------------------------------------
// Kernel 2: M-step. One wave per (n,x,y,c). Lanes: d = lane&15 (coalesced
// 64B vote rows), the two 16-lane halves split the 288 (o,i,j) terms.
// Single pass moments: mu = S1/S0, sigma = S2/S0 - mu^2.
// ---------------------------------------------------------------------------
__global__ __launch_bounds__(256) void cc_mstep(
    const float* __restrict__ votes, const float* __restrict__ Rp,
    const float* __restrict__ xin,
    float* __restrict__ muB, float* __restrict__ sigB, float* __restrict__ actB,
    float* __restrict__ out,
    const float* __restrict__ beta_v, const float* __restrict__ beta_a,
    const float* __restrict__ lambda_, int first, int last)
{
  int task = blockIdx.x * 8 + (threadIdx.x >> 5);   // 0..9215
  int lane = threadIdx.x & 31;
  int c = task % CCAP;
  int y = (task / CCAP) % WOUT;
  int x = (task / (CCAP*WOUT)) % WOUT;
  int n = task / (CCAP*NPOS);
  int d = lane & 15;
  int half = lane >> 4;

  float s0 = 0.f, s1 = 0.f, s2 = 0.f;
  for (int oij = half; oij < OIJ; oij += 2) {
    int o  = oij / 9;
    int ij = oij % 9;
    int i  = ij / 3, jj = ij % 3;
    size_t rbase = ((((size_t)n*BCAP + o)*3 + i)*3 + jj)*NPOS + x*WOUT + y;
    float rh = first ? (1.0f / (float)(NPOS*CCAP))
                     : Rp[rbase*CCAP + c];
    float av = xin[((size_t)n*CH_IN + 512 + o)*(WIN*WIN) + (2*x + i)*WIN + (2*y + jj)];
    rh *= av;
    float v = votes[(rbase*CCAP + c)*16 + d];
    s0 += rh;
    s1 += rh * v;
    s2 += rh * v * v;
  }
  s0 += __shfl_xor(s0, 16, 32);
  s1 += __shfl_xor(s1, 16, 32);
  s2 += __shfl_xor(s2, 16, 32);

  float mu = s1 / s0;
  float sg = fmaxf(s2 / s0 - mu * mu, 1e-30f);
  float cost = (beta_v[0] + logf(sg)) * s0;
  float cs = cost;
  cs += __shfl_xor(cs, 1, 32);
  cs += __shfl_xor(cs, 2, 32);
  cs += __shfl_xor(cs, 4, 32);
  cs += __shfl_xor(cs, 8, 32);
  float act = 1.0f / (1.0f + expf(-(lambda_[0] * (beta_a[0] - cs))));

  if (!last) {
    size_t mb = ((size_t)n*NPOS + x*WOUT + y)*CCAP + c;
    if (half == 0) { muB[mb*16 + d] = mu; sigB[mb*16 + d] = sg; }
    if (lane == 0) actB[mb] = act;
  } else {
    // mus: out[n, c*16+d, x, y] ; acts: out[n, 512+c, x, y]
    if (half == 0) out[((size_t)n*CH_IN + c*16 + d)*NPOS + x*WOUT + y] = mu;
    if (lane == 0) out[((size_t)n*CH_IN + 512 + c)*NPOS + x*WOUT + y] = act;
  }
}

// ---------------------------------------------------------------------------
// Kernel 3a: zero the window-overlap normalizer D.
// ---------------------------------------------------------------------------
__global__ __launch_bounds__(256) void cc_zero(float* __restrict__ p, int count)
{
  int i = blockIdx.x * 256 + threadIdx.x;
  if (i < count) p[i] = 0.0f;
}

// ---------------------------------------------------------------------------
// Kernel 3b: E-step. One wave per (n,o,i,j,x,y). Lanes: d = lane&15 for the
// Gaussian product (shuffle-xor product over 16 lanes); halves split c.
// votesE uses kperm = [0,2,1] applied to (i,j) when reading votes.
// Writes p_hat into Rp buffer, atomically accumulates T into D.
// ---------------------------------------------------------------------------
__global__ __launch_bounds__(256) void cc_estep(
    const float* __restrict__ votes,
    const float* __restrict__ muB, const float* __restrict__ sigB,
    const float* __restrict__ actB,
    float* __restrict__ Rp, float* __restrict__ Dacc)
{
  int task = blockIdx.x * 8 + (threadIdx.x >> 5);   // 0..82943
  int lane = threadIdx.x & 31;
  int y = task % WOUT;
  int x = (task / WOUT) % WOUT;
  int j = (task / NPOS) % 3;
  int i = (task / (NPOS*3)) % 3;
  int o = (task / (NPOS*9)) % BCAP;
  int n = task / (NPOS*9*BCAP);
  int d = lane & 15;
  int ch = lane >> 4;

  int ip = (3 - i) % 3;                 // kperm
  int jp = (3 - j) % 3;
  size_t vbase = ((((size_t)n*BCAP + o)*3 + ip)*3 + jp)*NPOS + x*WOUT + y;
  size_t rbase = ((((size_t)n*BCAP + o)*3 + i )*3 + j )*NPOS + x*WOUT + y;
  size_t mb = ((size_t)n*NPOS + x*WOUT + y)*CCAP;
  const float TWO_PI = 6.2831853071795864769f;

  float T = 0.f;
  for (int cc = 0; cc < CCAP; cc += 2) {
    int c = cc + ch;
    float vE = votes[(vbase*CCAP + c)*16 + d];
    float mu = muB[(mb + c)*16 + d];
    float sg = sigB[(mb + c)*16 + d];
    float df = vE - mu;
    float term = expf(-df*df) * rsqrtf(TWO_PI * sg);
    term *= __shfl_xor(term, 1, 32);
    term *= __shfl_xor(term, 2, 32);
    term *= __shfl_xor(term, 4, 32);
    term *= __shfl_xor(term, 8, 32);    // product over 16 d-lanes
    float ph = actB[mb + c] * term;
    if (d == 0) Rp[rbase*CCAP + c] = ph;
    T += ph;                            // each half-group sums its c subset
  }
  T += __shfl_xor(T, 16, 32);
  if (lane == 0)
    atomicAdd(&Dacc[((size_t)n*BCAP + o)*(WIN*WIN) + (2*x + i)*WIN + (2*y + j)], T);
}

// ---------------------------------------------------------------------------
// Kernel 3c: Rp = p_hat / D (gathered through the overlapping-window map).
// ---------------------------------------------------------------------------
__global__ __launch_bounds__(256) void cc_rnorm(
    float* __restrict__ Rp, const float* __restrict__ Dacc)
{
  size_t idx = (size_t)blockIdx.x * 256 + threadIdx.x;   // 0..2,654,207
  if (idx >= RP_N) return;
  size_t rest = idx / CCAP;
  int y = rest % WOUT; rest /= WOUT;
  int x = rest % WOUT; rest /= WOUT;
  int j = rest % 3;    rest /= 3;
  int i = rest % 3;    rest /= 3;
  int o = rest % BCAP;
  int n = (int)(rest / BCAP);
  float Dv = Dacc[((size_t)n*BCAP + o)*(WIN*WIN) + (2*x + i)*WIN + (2*y + j)];
  Rp[idx] = Rp[idx] / Dv;
}

// ---------------------------------------------------------------------------
extern "C" void kernel_launch(void* const* d_in, const int* in_sizes, int n_in,
                              void* d_out, int out_size, void* d_ws, size_t ws_size,
                              hipStream_t stream)
{
  const float* xin     = (const float*)d_in[0];   // (8,544,13,13)
  const float* Wt      = (const float*)d_in[1];   // (3,3,32,32,4,4)
  const float* beta_v  = (const float*)d_in[2];
  const float* beta_a  = (const float*)d_in[3];
  const float* lambda_ = (const float*)d_in[4];
  float* out = (float*)d_out;                     // (8,544,6,6)

  float* votes = (float*)d_ws;
  float* Rp    = votes + VOTES_N;
  float* muB   = Rp + RP_N;
  float* sigB  = muB + MU_N;
  float* actB  = sigB + MU_N;
  float* Dacc  = actB + ACT_N;

  // votes: 2304 blocks x 8 waves, each wave does 9 WMMA f32 16x16x4
  cc_votes_wmma<<<NB*BCAP*KSZ*KSZ, 256, 0, stream>>>(xin, Wt, votes);

  for (int it = 0; it < 3; ++it) {
    int first = (it == 0);
    int last  = (it == 2);
    cc_mstep<<<(NB*NPOS*CCAP)/8, 256, 0, stream>>>(
        votes, Rp, xin, muB, sigB, actB, out, beta_v, beta_a, lambda_, first, last);
    if (!last) {
      cc_zero<<<((int)D_N + 255)/256, 256, 0, stream>>>(Dacc, (int)D_N);
      cc_estep<<<(NB*BCAP*KSZ*KSZ*NPOS)/8, 256, 0, stream>>>(
          votes, muB, sigB, actB, Rp, Dacc);
      cc_rnorm<<<(int)((RP_N + 255)/256), 256, 0, stream>>>(Rp, Dacc);
    }
  }
}